// SparseLinear_68556267978819
// MI455X (gfx1250) — compile-verified
//
#include <hip/hip_runtime.h>

#ifndef __has_builtin
#define __has_builtin(x) 0
#endif

#define WAVES_PER_BLOCK 8
#define CHUNK 32   // nonzeros staged per wave per iteration (one per lane)

#if __has_builtin(__builtin_amdgcn_global_load_async_to_lds_b32)
#define HAVE_ASYNC_LDS 1
#else
#define HAVE_ASYNC_LDS 0
#endif

// Cast any pointer to an address-space(1)/(3) int* in two steps:
// C-style cast to generic int* (strips const / reinterprets float*),
// then an addrspacecast to the AS-qualified pointer type.
typedef __attribute__((address_space(1))) int gas_int;
typedef __attribute__((address_space(3))) int las_int;
#define GAS_PTR(p) ((gas_int*)(int*)(p))
#define LAS_PTR(p) ((las_int*)(int*)(p))

template <int N>
__device__ __forceinline__ void wait_async_le() {
#if __has_builtin(__builtin_amdgcn_s_wait_asynccnt)
  __builtin_amdgcn_s_wait_asynccnt(N);
#else
  if (N == 0)      asm volatile("s_wait_asynccnt 0x0" ::: "memory");
  else if (N == 3) asm volatile("s_wait_asynccnt 0x3" ::: "memory");
  else             asm volatile("s_wait_asynccnt 0x0" ::: "memory");
#endif
#if __has_builtin(__builtin_amdgcn_wave_barrier)
  __builtin_amdgcn_wave_barrier();
#endif
  asm volatile("" ::: "memory");
}

// Stage one chunk of 32 (row, col, val) triples into this wave's LDS buffers.
// Each lane moves one 4B element of each array -> 3 coalesced 128B async copies.
__device__ __forceinline__ void stage_chunk(const int* __restrict__ rows,
                                            const int* __restrict__ cols,
                                            const float* __restrict__ vals,
                                            int base, int lane,
                                            unsigned* sr, unsigned* sc, float* sv) {
#if HAVE_ASYNC_LDS
  __builtin_amdgcn_global_load_async_to_lds_b32(GAS_PTR(rows + base + lane),
                                                LAS_PTR(sr + lane), 0, 0);
  __builtin_amdgcn_global_load_async_to_lds_b32(GAS_PTR(cols + base + lane),
                                                LAS_PTR(sc + lane), 0, 0);
  __builtin_amdgcn_global_load_async_to_lds_b32(GAS_PTR(vals + base + lane),
                                                LAS_PTR(sv + lane), 0, 0);
#else
  sr[lane] = (unsigned)rows[base + lane];
  sc[lane] = (unsigned)cols[base + lane];
  sv[lane] = vals[base + lane];
#if __has_builtin(__builtin_amdgcn_wave_barrier)
  __builtin_amdgcn_wave_barrier();
#endif
#endif
}

// Consume one staged chunk: every lane handles 4 of the 128 output columns.
// x-row gather is one coalesced 512B load per nonzero; scatter is 4 fp32
// global atomics per lane (L2-resident since out is only 4MB).
__device__ __forceinline__ void process_chunk(const unsigned* sr, const unsigned* sc,
                                              const float* sv,
                                              const float* __restrict__ x,
                                              float* __restrict__ out,
                                              int lane) {
#pragma unroll 4
  for (int j = 0; j < CHUNK; ++j) {
    const unsigned row = sr[j];           // LDS broadcast read (uniform address)
    const unsigned col = sc[j];
    const float    v   = sv[j];
    const float4 xv = *((const float4*)(x + ((size_t)col << 7)) + lane);
    float* o = out + ((size_t)row << 7) + (lane << 2);
    unsafeAtomicAdd(o + 0, v * xv.x);     // global_atomic_add_f32 (no return)
    unsafeAtomicAdd(o + 1, v * xv.y);
    unsafeAtomicAdd(o + 2, v * xv.z);
    unsafeAtomicAdd(o + 3, v * xv.w);
  }
}

__global__ __launch_bounds__(256) void SparseLinear_scatter_kernel(
    const int* __restrict__ rows, const int* __restrict__ cols,
    const float* __restrict__ vals, const float* __restrict__ x,
    float* __restrict__ out, int nnz) {
  __shared__ unsigned s_rows[2][WAVES_PER_BLOCK][CHUNK];
  __shared__ unsigned s_cols[2][WAVES_PER_BLOCK][CHUNK];
  __shared__ float    s_vals[2][WAVES_PER_BLOCK][CHUNK];

  const int lane = threadIdx.x & 31;
  const int wid  = threadIdx.x >> 5;                     // wave in block (wave32)
  const int gw   = blockIdx.x * WAVES_PER_BLOCK + wid;   // global wave id
  const int nw   = gridDim.x * WAVES_PER_BLOCK;          // total waves
  const int nChunks = nnz / CHUNK;

  int c = gw;
  int buf = 0;
  if (c < nChunks)
    stage_chunk(rows, cols, vals, c * CHUNK, lane,
                s_rows[0][wid], s_cols[0][wid], s_vals[0][wid]);

  for (; c < nChunks; c += nw) {
    const int nc = c + nw;
    if (nc < nChunks) {
      stage_chunk(rows, cols, vals, nc * CHUNK, lane,
                  s_rows[buf ^ 1][wid], s_cols[buf ^ 1][wid], s_vals[buf ^ 1][wid]);
      wait_async_le<3>();   // in-order retire: current buffer's 3 copies are done
    } else {
      wait_async_le<0>();   // last chunk: drain everything
    }
    process_chunk(s_rows[buf][wid], s_cols[buf][wid], s_vals[buf][wid], x, out, lane);
    buf ^= 1;
  }

  // Tail (nnz % 32) handled by wave 0 of block 0 with direct loads.
  const int tail = nnz - nChunks * CHUNK;
  if (tail && blockIdx.x == 0 && wid == 0) {
    const int base = nChunks * CHUNK;
    for (int t = 0; t < tail; ++t) {
      const unsigned row = (unsigned)rows[base + t];
      const unsigned col = (unsigned)cols[base + t];
      const float    v   = vals[base + t];
      const float4 xv = *((const float4*)(x + ((size_t)col << 7)) + lane);
      float* o = out + ((size_t)row << 7) + (lane << 2);
      unsafeAtomicAdd(o + 0, v * xv.x);
      unsafeAtomicAdd(o + 1, v * xv.y);
      unsafeAtomicAdd(o + 2, v * xv.z);
      unsafeAtomicAdd(o + 3, v * xv.w);
    }
  }
}

// One float4 store per thread; grid sized exactly by the host launcher.
__global__ __launch_bounds__(256) void SparseLinear_zero_kernel(float4* __restrict__ out, int n4) {
  const int i = blockIdx.x * blockDim.x + threadIdx.x;
  if (i < n4) out[i] = make_float4(0.f, 0.f, 0.f, 0.f);
}

extern "C" void kernel_launch(void* const* d_in, const int* in_sizes, int n_in,
                              void* d_out, int out_size, void* d_ws, size_t ws_size,
                              hipStream_t stream) {
  const int*   ids  = (const int*)d_in[0];    // [2, nnz]: rows then cols
  const float* vals = (const float*)d_in[1];  // [nnz]
  const float* x    = (const float*)d_in[2];  // [8192, 128]
  float* out = (float*)d_out;                 // [8192, 128]

  const int nnz  = in_sizes[1];
  const int* rows = ids;
  const int* cols = ids + nnz;

  // Must zero every launch: output is built with atomic accumulation.
  const int n4 = out_size >> 2;               // out_size is a multiple of 128
  SparseLinear_zero_kernel<<<(n4 + 255) / 256, 256, 0, stream>>>((float4*)out, n4);

  // 1024 blocks x 8 waves = 8192 waves; 65536 chunks -> 8 chunks/wave,
  // double-buffered through LDS with async global->LDS copies.
  SparseLinear_scatter_kernel<<<1024, 256, 0, stream>>>(rows, cols, vals, x, out, nnz);
}